// HexPlaneField_41420664603263
// MI455X (gfx1250) — compile-verified
//
#include <hip/hip_runtime.h>
#include <hip/hip_bf16.h>
#include <stdint.h>

#define NPLANES 24

struct Planes { const float* p[NPLANES]; };

struct TransDesc {
  const float* src;   // (32, H, W)
  float*       dst;   // (H, W, 32)
  int W;              // row length
  int HW;             // H*W (channel stride)
  int bstart;         // first block id of this plane
  int wtiles;         // W/32
};
struct TransTable { TransDesc d[NPLANES]; };

// ---------------- CDNA5 async global->LDS helpers (guarded) ----------------
// cpol=1 -> TH=NT (gfx12+ CPol: TH in bits[2:0]); source is read-once, keep it
// from evicting the L2-resident transposed grids we are producing.
__device__ __forceinline__ void cdna5_async_g2l_b32(const float* gsrc, float* ldst) {
#if defined(__gfx1250__) && __has_builtin(__builtin_amdgcn_global_load_async_to_lds_b32)
  __builtin_amdgcn_global_load_async_to_lds_b32(
      (__attribute__((address_space(1))) int*)gsrc,
      (__attribute__((address_space(3))) int*)ldst,
      /*offset=*/0, /*cpol=*/1);
#else
  *ldst = __builtin_nontemporal_load(gsrc);  // synchronous fallback
#endif
}

__device__ __forceinline__ void cdna5_wait_async0() {
#if defined(__gfx1250__)
#if __has_builtin(__builtin_amdgcn_s_wait_asynccnt)
  __builtin_amdgcn_s_wait_asynccnt(0);
#else
  asm volatile("s_wait_asynccnt 0" ::: "memory");
#endif
#endif
}

// -------- Single-launch transpose of ALL planes (C,H,W) -> (H,W,C) ---------
// One block = one 32x32 tile (32 x-positions of one row y, all 32 channels).
// Load phase : lane = x  (coalesced NT reads, async copy into LDS)
// Store phase: lane = c  (coalesced 128B RT writes -> stays in L2 for gather)
__global__ __launch_bounds__(256) void transpose_all(TransTable tt)
{
  __shared__ float tile[32][33];           // +1 pad: conflict-free both phases
  const int bid = blockIdx.x;

  // Uniform (per-block) plane lookup: tiny SALU scan over 24 descriptors.
  int i = 0;
#pragma unroll 1
  for (int k = 1; k < NPLANES; ++k)
    if (bid >= tt.d[k].bstart) i = k;
  const TransDesc d = tt.d[i];

  const int local = bid - d.bstart;
  const int ty    = local / d.wtiles;      // row y
  const int xb    = (local - ty * d.wtiles) * 32;

  const int lx = threadIdx.x;              // 0..31
  const int ly = threadIdx.y;              // 0..7

  const float* srow = d.src + ty * d.W + xb + lx;
#pragma unroll
  for (int cc = 0; cc < 4; ++cc) {
    const int c = ly + 8 * cc;             // channel 0..31
    cdna5_async_g2l_b32(srow + c * d.HW, &tile[c][lx]);
  }
  cdna5_wait_async0();
  __syncthreads();

  float* drow = d.dst + (ty * d.W + xb) * 32 + lx;  // lane = channel now
#pragma unroll
  for (int xx = 0; xx < 4; ++xx) {
    const int xl = ly + 8 * xx;            // x within tile
    drow[xl * 32] = tile[lx][xl];          // RT: park in L2 for the gather
  }
}

// -------- Main gather kernel: one wave32 per point, lane = channel ---------
template <bool TRANSPOSED>
__global__ __launch_bounds__(256) void hexplane_gather(
    const float* __restrict__ pts, const float* __restrict__ tmv,
    Planes pl, float* __restrict__ out, int npts)
{
  const int tid  = blockIdx.x * 256 + threadIdx.x;
  const int pt   = tid >> 5;
  const int lane = tid & 31;
  if (pt >= npts) return;

  // Read-once inputs: non-temporal so they never displace grid data in L2.
  float coord[4];
  coord[0] = -__builtin_nontemporal_load(&pts[3 * pt + 0]) / 1.6f;
  coord[1] = -__builtin_nontemporal_load(&pts[3 * pt + 1]) / 1.6f;
  coord[2] = -__builtin_nontemporal_load(&pts[3 * pt + 2]) / 1.6f;
  coord[3] = __builtin_nontemporal_load(&tmv[pt]);

  float* o = out + (size_t)pt * 128 + lane;

#pragma unroll
  for (int s = 0; s < 4; ++s) {
    const int m = 1 << s;
    // Per-axis interpolation tuples (shared by the 6 planes of this scale).
    int i0[4], i1[4];
    float fr[4];
#pragma unroll
    for (int d = 0; d < 4; ++d) {
      const int R = (d < 3) ? (64 * m) : 150;
      float x = (coord[d] + 1.0f) * (0.5f * (float)(R - 1));
      x = fminf(fmaxf(x, 0.0f), (float)(R - 1));
      const float xf = floorf(x);
      const int a = (int)xf;
      int b = a + 1; if (b > R - 1) b = R - 1;
      i0[d] = a; i1[d] = b; fr[d] = x - xf;
    }

    const int c0s[6] = {0, 0, 0, 1, 1, 2};
    const int c1s[6] = {1, 2, 3, 2, 3, 3};
    float acc = 1.0f;
#pragma unroll
    for (int p = 0; p < 6; ++p) {
      const int a = c0s[p], b = c1s[p];
      const int W = 64 * m;                        // reso[a], a < 3 always
      const int H = (b < 3) ? (64 * m) : 150;
      const float* __restrict__ g = pl.p[s * 6 + p];
      const int x0 = i0[a], x1 = i1[a], y0 = i0[b], y1 = i1[b];
      const float wx = fr[a], wy = fr[b];
      float g00, g01, g10, g11;
      if (TRANSPOSED) {
        // (H, W, 32) layout: lanes read 32 consecutive floats -> coalesced,
        // default RT policy: grids (~189 MB) live in the 192 MB L2.
        g00 = g[(y0 * W + x0) * 32 + lane];
        g01 = g[(y0 * W + x1) * 32 + lane];
        g10 = g[(y1 * W + x0) * 32 + lane];
        g11 = g[(y1 * W + x1) * 32 + lane];
      } else {
        const int cb = lane * H * W;               // (32, H, W) layout
        g00 = g[cb + y0 * W + x0];
        g01 = g[cb + y0 * W + x1];
        g10 = g[cb + y1 * W + x0];
        g11 = g[cb + y1 * W + x1];
      }
      const float top = g00 * (1.0f - wx) + g01 * wx;
      const float bot = g10 * (1.0f - wx) + g11 * wx;
      acc *= top * (1.0f - wy) + bot * wy;
    }
    // Output (256 MB, write-once): non-temporal store, coalesced 128B/wave,
    // streams to HBM without flushing the grid working set out of L2.
    __builtin_nontemporal_store(acc, &o[32 * s]);
  }
}

// ---------------------------------------------------------------------------
extern "C" void kernel_launch(void* const* d_in, const int* in_sizes, int n_in,
                              void* d_out, int out_size, void* d_ws, size_t ws_size,
                              hipStream_t stream) {
  (void)n_in; (void)out_size;
  const float* pts = (const float*)d_in[0];
  const float* tmv = (const float*)d_in[1];
  const int npts = in_sizes[0] / 3;

  static const int c0s[6] = {0, 0, 0, 1, 1, 2};
  static const int c1s[6] = {1, 2, 3, 2, 3, 3};
  int Ws[NPLANES], Hs[NPLANES];
  size_t offs[NPLANES], total = 0;
  for (int s = 0; s < 4; ++s) {
    const int m = 1 << s;
    const int reso[4] = {64 * m, 64 * m, 64 * m, 150};
    for (int p = 0; p < 6; ++p) {
      const int i = s * 6 + p;
      Ws[i] = reso[c0s[p]];
      Hs[i] = reso[c1s[p]];
      offs[i] = total;
      total += (size_t)Ws[i] * Hs[i] * 32;
    }
  }

  const int blocks = (npts + 7) / 8;   // 8 wave32 points per 256-thread block
  Planes pl;

  if (ws_size >= total * sizeof(float)) {
    float* ws = (float*)d_ws;
    TransTable tt;
    int nb = 0;
    for (int i = 0; i < NPLANES; ++i) {
      tt.d[i].src    = (const float*)d_in[2 + i];
      tt.d[i].dst    = ws + offs[i];
      tt.d[i].W      = Ws[i];
      tt.d[i].HW     = Hs[i] * Ws[i];
      tt.d[i].bstart = nb;
      tt.d[i].wtiles = Ws[i] / 32;     // W always a multiple of 32
      nb += (Ws[i] / 32) * Hs[i];
      pl.p[i] = ws + offs[i];
    }
    transpose_all<<<nb, dim3(32, 8), 0, stream>>>(tt);
    hexplane_gather<true><<<blocks, 256, 0, stream>>>(pts, tmv, pl,
                                                      (float*)d_out, npts);
  } else {
    for (int i = 0; i < NPLANES; ++i) pl.p[i] = (const float*)d_in[2 + i];
    hexplane_gather<false><<<blocks, 256, 0, stream>>>(pts, tmv, pl,
                                                       (float*)d_out, npts);
  }
}